// Critic_55997783605448
// MI455X (gfx1250) — compile-verified
//
#include <hip/hip_runtime.h>
#include <hip/hip_bf16.h>

typedef __attribute__((ext_vector_type(2))) float v2f;
typedef __attribute__((ext_vector_type(8))) float v8f;

// ---------------- utility ----------------
__global__ void zero_f32(float* __restrict__ p, int n) {
  int i = blockIdx.x * blockDim.x + threadIdx.x;
  if (i < n) p[i] = 0.0f;
}

// deg[dst[e]] += 1
__global__ void degree_kernel(const int* __restrict__ dst, float* __restrict__ deg, int E) {
  int e = blockIdx.x * blockDim.x + threadIdx.x;
  if (e < E) atomicAdd(&deg[dst[e]], 1.0f);
}

__global__ void dinv_kernel(const float* __restrict__ deg, float* __restrict__ dinv, int n) {
  int i = blockIdx.x * blockDim.x + threadIdx.x;
  if (i < n) dinv[i] = rsqrtf(1.0f + deg[i]);
}

// ---------------- WMMA GEMM: C[M x Ncols] = A[M x 64] * B[64 x Ncols] ----------------
// One wave32 computes a 16x64 output tile using v_wmma_f32_16x16x4_f32.
// Requires M % 16 == 0 and Ncols % 64 == 0 (true here: 100000, 256; 64, 256).
__global__ void gemm_k64_wmma(const float* __restrict__ A, const float* __restrict__ B,
                              float* __restrict__ C, int M, int Ncols,
                              const float* __restrict__ bias, int do_relu) {
  int wave = threadIdx.x >> 5;
  int lane = threadIdx.x & 31;
  int rt = blockIdx.x * (blockDim.x >> 5) + wave;   // row-tile index (wave-uniform)
  if (rt * 16 >= M) return;                          // uniform exit: EXEC stays all-1s
  int cb = blockIdx.y * 64;                          // column base
  int m  = lane & 15;                                // row-in-tile / col-in-tile
  int hh = lane >> 4;                                // lane half selects K parity pair

  v8f c0 = {}, c1 = {}, c2 = {}, c3 = {};

  // A layout (16x4 f32): lane holds M=lane%16; VGPR v + 2*half -> K
  const float* arow = A + ((size_t)(rt * 16 + m)) * 64 + 2 * hh;
  // B layout (4x16 f32): lane holds N=lane%16; VGPR v + 2*half -> K
  const float* bcol = B + cb + m + (size_t)(2 * hh) * Ncols;

#pragma unroll
  for (int kk = 0; kk < 64; kk += 4) {
    float2 av = *(const float2*)(arow + kk);
    v2f a; a.x = av.x; a.y = av.y;
    const float* bp = bcol + (size_t)kk * Ncols;
    v2f b0, b1v, b2v, b3v;
    b0.x  = bp[0];   b0.y  = bp[Ncols];
    b1v.x = bp[16];  b1v.y = bp[Ncols + 16];
    b2v.x = bp[32];  b2v.y = bp[Ncols + 32];
    b3v.x = bp[48];  b3v.y = bp[Ncols + 48];
    c0 = __builtin_amdgcn_wmma_f32_16x16x4_f32(false, a, false, b0,  (short)0, c0, false, false);
    c1 = __builtin_amdgcn_wmma_f32_16x16x4_f32(false, a, false, b1v, (short)0, c1, false, false);
    c2 = __builtin_amdgcn_wmma_f32_16x16x4_f32(false, a, false, b2v, (short)0, c2, false, false);
    c3 = __builtin_amdgcn_wmma_f32_16x16x4_f32(false, a, false, b3v, (short)0, c3, false, false);
  }

  float bv0 = 0.f, bv1 = 0.f, bv2 = 0.f, bv3 = 0.f;
  if (bias) {
    bv0 = bias[cb + m];
    bv1 = bias[cb + 16 + m];
    bv2 = bias[cb + 32 + m];
    bv3 = bias[cb + 48 + m];
  }

  // C/D layout: VGPR v holds row v (lanes 0-15) / v+8 (lanes 16-31), col = lane%16
#pragma unroll
  for (int v = 0; v < 8; ++v) {
    int row = rt * 16 + v + 8 * hh;
    float* crow = C + (size_t)row * Ncols + cb + m;
    float o0 = c0[v] + bv0, o1 = c1[v] + bv1, o2 = c2[v] + bv2, o3 = c3[v] + bv3;
    if (do_relu) {
      o0 = fmaxf(o0, 0.f); o1 = fmaxf(o1, 0.f);
      o2 = fmaxf(o2, 0.f); o3 = fmaxf(o3, 0.f);
    }
    crow[0] = o0; crow[16] = o1; crow[32] = o2; crow[48] = o3;
  }
}

// ---------------- edge scatter: agg[dst] += h[src] * dinv[src]*dinv[dst] ----------------
// One wave32 per edge; each lane owns a float2 of the 64-float feature row.
__global__ void scatter_kernel(const float* __restrict__ h, const float* __restrict__ dinv,
                               const int* __restrict__ src, const int* __restrict__ dst,
                               float* __restrict__ agg, int E) {
  int e = blockIdx.x * (blockDim.x >> 5) + (threadIdx.x >> 5);
  if (e >= E) return;
  int lane = threadIdx.x & 31;
  int s = src[e], d = dst[e];
  float w = dinv[s] * dinv[d];
  float2 v = *(const float2*)(h + (size_t)s * 64 + 2 * lane);
  float* out = agg + (size_t)d * 64 + 2 * lane;
  atomicAdd(out,     v.x * w);
  atomicAdd(out + 1, v.y * w);
}

// x_next = relu(agg + h * dinv^2 + bias)   (written in-place over agg)
__global__ void finalize_kernel(const float* __restrict__ agg, const float* __restrict__ h,
                                const float* __restrict__ dinv, const float* __restrict__ bias,
                                float* __restrict__ xout, int n) {
  int t = blockIdx.x * blockDim.x + threadIdx.x;
  if (t >= n * 64) return;
  int i = t >> 6, f = t & 63;
  float di = dinv[i];
  float val = agg[t] + h[t] * (di * di) + bias[f];
  xout[t] = fmaxf(val, 0.f);
}

// ---------------- pooling ----------------
__global__ void pool_accum(const float* __restrict__ x, const int* __restrict__ batch,
                           float* __restrict__ sums, float* __restrict__ cnt, int n) {
  int i = blockIdx.x * (blockDim.x >> 5) + (threadIdx.x >> 5);
  if (i >= n) return;
  int lane = threadIdx.x & 31;
  int g = batch[i];
  float2 v = *(const float2*)(x + (size_t)i * 64 + 2 * lane);
  float* out = sums + (size_t)g * 64 + 2 * lane;
  atomicAdd(out,     v.x);
  atomicAdd(out + 1, v.y);
  if (lane == 0) atomicAdd(&cnt[g], 1.0f);
}

__global__ void pool_div(const float* __restrict__ sums, const float* __restrict__ cnt,
                         float* __restrict__ pooled, int total) {
  int t = blockIdx.x * blockDim.x + threadIdx.x;
  if (t >= total) return;
  int g = t >> 6;
  pooled[t] = sums[t] / fmaxf(cnt[g], 1.0f);
}

// ---------------- tiny final GEMM: C[M x N] = A[M x K] * B[K x N] + bias ----------------
__global__ void gemm_small(const float* __restrict__ A, const float* __restrict__ B,
                           const float* __restrict__ bias, float* __restrict__ C,
                           int M, int K, int N) {
  int t = blockIdx.x * blockDim.x + threadIdx.x;
  if (t >= M * N) return;
  int i = t / N, j = t % N;
  float s = bias[j];
  for (int k = 0; k < K; ++k) s += A[(size_t)i * K + k] * B[(size_t)k * N + j];
  C[t] = s;
}

extern "C" void kernel_launch(void* const* d_in, const int* in_sizes, int n_in,
                              void* d_out, int out_size, void* d_ws, size_t ws_size,
                              hipStream_t stream) {
  (void)n_in; (void)out_size; (void)ws_size;
  const float* x   = (const float*)d_in[0];   // [N,64]
  const float* cW  = (const float*)d_in[1];   // [3,64,64]
  const float* cbv = (const float*)d_in[2];   // [3,64]
  const float* W1  = (const float*)d_in[3];   // [64,256]
  const float* b1  = (const float*)d_in[4];   // [256]
  const float* W2  = (const float*)d_in[5];   // [256,10]
  const float* b2  = (const float*)d_in[6];   // [10]
  const int*   ei  = (const int*)d_in[7];     // [2,E]
  const int*   bat = (const int*)d_in[8];     // [N]

  const int N = in_sizes[0] / 64;
  const int E = in_sizes[7] / 2;
  const int G = 256;   // graphs
  const int F = 64;    // feature dim
  const int H = 256;   // hidden
  const int L = 10;    // labels

  // workspace layout (floats)
  float* ws     = (float*)d_ws;
  float* deg    = ws;                              // N
  float* dinv   = deg  + N;                        // N
  float* bufH   = dinv + N;                        // N*64
  float* bufA   = bufH + (size_t)N * F;            // N*64
  float* bufB   = bufA + (size_t)N * F;            // N*64
  float* sums   = bufB + (size_t)N * F;            // G*64
  float* cnt    = sums + (size_t)G * F;            // G
  float* pooled = cnt  + G;                        // G*64
  float* h1     = pooled + (size_t)G * F;          // G*H

  const int T = 256; // threads per block (8 waves)

  // degree + normalization
  zero_f32<<<(N + T - 1) / T, T, 0, stream>>>(deg, N);
  degree_kernel<<<(E + T - 1) / T, T, 0, stream>>>(ei + E, deg, E);
  dinv_kernel<<<(N + T - 1) / T, T, 0, stream>>>(deg, dinv, N);

  // 3 GCN layers
  const float* xcur = x;
  float* aggs[3] = { bufA, bufB, bufA };
  int rowTiles = N / 16;                // 6250 (N multiple of 16)
  int gemmBlocks = (rowTiles + 7) / 8;  // 8 waves per block
  for (int l = 0; l < 3; ++l) {
    float* agg = aggs[l];
    gemm_k64_wmma<<<dim3(gemmBlocks, 1), T, 0, stream>>>(
        xcur, cW + (size_t)l * F * F, bufH, N, F, nullptr, 0);
    zero_f32<<<((N * F) + T - 1) / T, T, 0, stream>>>(agg, N * F);
    scatter_kernel<<<(E + 7) / 8, T, 0, stream>>>(bufH, dinv, ei, ei + E, agg, E);
    finalize_kernel<<<((N * F) + T - 1) / T, T, 0, stream>>>(
        agg, bufH, dinv, cbv + (size_t)l * F, agg, N);
    xcur = agg;
  }

  // global mean pool
  zero_f32<<<((G * F) + T - 1) / T, T, 0, stream>>>(sums, G * F);
  zero_f32<<<1, T, 0, stream>>>(cnt, G);
  pool_accum<<<(N + 7) / 8, T, 0, stream>>>(xcur, bat, sums, cnt, N);
  pool_div<<<((G * F) + T - 1) / T, T, 0, stream>>>(sums, cnt, pooled, G * F);

  // MLP: h1 = relu(pooled @ W1 + b1)  -> 256x64 @ 64x256 via WMMA
  gemm_k64_wmma<<<dim3((G / 16 + 7) / 8, H / 64), T, 0, stream>>>(
      pooled, W1, h1, G, H, b1, 1);

  // out = h1 @ W2 + b2  -> 256x256 @ 256x10 (tiny)
  gemm_small<<<((G * L) + T - 1) / T, T, 0, stream>>>(h1, W2, b2, (float*)d_out, G, H, L);
}